// ScoringLayer_57312043598524
// MI455X (gfx1250) — compile-verified
//
#include <hip/hip_runtime.h>

#define BATCH      4096
#define FILTER_NUM 1024
#define REL_CNT    19
#define REL_DIM    1024
#define SLAB       (FILTER_NUM * REL_CNT)   // 19456 floats = 77824 bytes per b

typedef float v2f __attribute__((ext_vector_type(2)));
typedef float v8f __attribute__((ext_vector_type(8)));
typedef int   gi4 __attribute__((vector_size(16)));   // matches builtin param type

#define AS1 __attribute__((address_space(1)))
#define AS3 __attribute__((address_space(3)))

#if defined(__has_builtin)
#  if __has_builtin(__builtin_amdgcn_global_load_async_to_lds_b128) && \
      __has_builtin(__builtin_amdgcn_s_wait_asynccnt)
#    define USE_ASYNC_LDS 1
#  endif
#endif

// ---------------------------------------------------------------------------
// Stage 1: V[f][r] = sum_e U[f][e] * R[r][e]   (1024x19, K=1024)
// One wave32 per 16x16 output tile, V_WMMA_F32_16X16X4_F32 over K in steps of 4.
// A (16x4 f32): lane<16 -> M=lane, K={0,1}; lane>=16 -> M=lane-16, K={2,3}.
// B (4x16 f32): mirrored: lane<16 -> N=lane, K={0,1}; lane>=16 -> K={2,3}.
// D (16x16 f32, 8 VGPRs): element (M = v + 8*(lane>=16), N = lane%16).
// ---------------------------------------------------------------------------
__global__ __launch_bounds__(32)
void uproj_wmma_kernel(const float* __restrict__ U,
                       const float* __restrict__ Rm,
                       float* __restrict__ V) {
    const int lane = threadIdx.x;        // 0..31
    const int half = lane >> 4;          // 0 or 1 (selects K pair 0/1 vs 2/3)
    const int sub  = lane & 15;
    const int m0   = blockIdx.x * 16;    // f tile
    const int n0   = blockIdx.y * 16;    // r tile (19 -> 2 tiles with guard)

    const int  row   = m0 + sub;                       // f index for A
    const int  col   = n0 + sub;                       // r index for B
    const bool colok = (col < REL_CNT);
    const int  colc  = colok ? col : (REL_CNT - 1);    // clamp, zero data below

    const float* arow = U  + (size_t)row  * REL_DIM + 2 * half;
    const float* brow = Rm + (size_t)colc * REL_DIM + 2 * half;

    v8f c = {0.f, 0.f, 0.f, 0.f, 0.f, 0.f, 0.f, 0.f};

    for (int k = 0; k < REL_DIM; k += 4) {
        v2f a = *(const v2f*)(arow + k);
        v2f b = *(const v2f*)(brow + k);
        if (!colok) { b.x = 0.f; b.y = 0.f; }
        // (neg_a, A, neg_b, B, c_mod, C, reuse_a, reuse_b)
        c = __builtin_amdgcn_wmma_f32_16x16x4_f32(
                false, a, false, b, (short)0, c, false, false);
    }

    if (colok) {
        #pragma unroll
        for (int v = 0; v < 8; ++v) {
            int m = m0 + v + 8 * half;
            V[(size_t)m * REL_CNT + col] = c[v];
        }
    }
}

// ---------------------------------------------------------------------------
// Stage 2: scores[b][r] = sum_f conv[b][f][r] * V[f][r]
// One block per b. Stage the 76KB conv[b] slab into LDS with async b128 loads
// (coalesced full cachelines), then each thread reduces a contiguous 76-elem
// chunk (76 = 4*19, chunk start = 76*t = 0 mod 19 -> compile-time r indices).
// ---------------------------------------------------------------------------
__global__ __launch_bounds__(256)
void score_kernel(const float* __restrict__ conv,
                  const float* __restrict__ V,
                  float* __restrict__ out) {
    __shared__ __align__(16) float slab[SLAB];       // 77824 B (LDS = 320KB/WGP)
    __shared__ float wred[8][REL_CNT];

    const int b = blockIdx.x;
    const int t = threadIdx.x;
    const float* gsrc = conv + (size_t)b * SLAB;

    // ---- stage slab into LDS: 4864 float4 = 256 threads x 19 iterations ----
#if defined(USE_ASYNC_LDS)
    #pragma unroll
    for (int k = 0; k < 19; ++k) {
        int f4 = k * 256 + t;                         // consecutive lanes -> consecutive 16B
        __builtin_amdgcn_global_load_async_to_lds_b128(
            (AS1 gi4*)(gsrc + (size_t)f4 * 4),
            (AS3 gi4*)(slab + (size_t)f4 * 4),
            0, 0);
    }
    __builtin_amdgcn_s_wait_asynccnt(0);
    __syncthreads();
#else
    #pragma unroll
    for (int k = 0; k < 19; ++k) {
        int f4 = k * 256 + t;
        ((float4*)slab)[f4] = ((const float4*)gsrc)[f4];
    }
    __syncthreads();
#endif

    // ---- per-thread partial dot over its 76-element chunk ----
    float acc[REL_CNT];
    #pragma unroll
    for (int r = 0; r < REL_CNT; ++r) acc[r] = 0.f;

    const float4* lsrc = (const float4*)(slab + t * 76);   // 304 B, 16B aligned
    const float4* vsrc = (const float4*)(V    + t * 76);   // same flat layout

    #pragma unroll
    for (int m = 0; m < 19; ++m) {
        float4 x = lsrc[m];
        float4 v = vsrc[m];
        acc[(4 * m + 0) % REL_CNT] += x.x * v.x;
        acc[(4 * m + 1) % REL_CNT] += x.y * v.y;
        acc[(4 * m + 2) % REL_CNT] += x.z * v.z;
        acc[(4 * m + 3) % REL_CNT] += x.w * v.w;
    }

    // ---- wave32 shuffle reduction of the 19 accumulators ----
    #pragma unroll
    for (int r = 0; r < REL_CNT; ++r) {
        float a = acc[r];
        a += __shfl_down(a, 16, 32);
        a += __shfl_down(a,  8, 32);
        a += __shfl_down(a,  4, 32);
        a += __shfl_down(a,  2, 32);
        a += __shfl_down(a,  1, 32);
        acc[r] = a;
    }

    const int wave = t >> 5;
    const int lane = t & 31;
    if (lane == 0) {
        #pragma unroll
        for (int r = 0; r < REL_CNT; ++r) wred[wave][r] = acc[r];
    }
    __syncthreads();

    if (t < REL_CNT) {
        float s = 0.f;
        #pragma unroll
        for (int w = 0; w < 8; ++w) s += wred[w][t];
        out[(size_t)b * REL_CNT + t] = s;
    }
}

extern "C" void kernel_launch(void* const* d_in, const int* in_sizes, int n_in,
                              void* d_out, int out_size, void* d_ws, size_t ws_size,
                              hipStream_t stream) {
    const float* conv = (const float*)d_in[0];  // [4096,1024,19]
    const float* Rm   = (const float*)d_in[1];  // [19,1024]
    const float* U    = (const float*)d_in[2];  // [1024,1024]
    float* out = (float*)d_out;                 // [4096,19]
    float* V   = (float*)d_ws;                  // [1024,19] scratch

    (void)in_sizes; (void)n_in; (void)out_size; (void)ws_size;

    // Stage 1: V = U @ R^T via WMMA (64 f-tiles x 2 r-tiles, 1 wave each)
    uproj_wmma_kernel<<<dim3(64, 2, 1), 32, 0, stream>>>(U, Rm, V);

    // Stage 2: stream conv once (318 MB -> ~14us @ 23.3 TB/s), reduce per r
    score_kernel<<<dim3(BATCH, 1, 1), 256, 0, stream>>>(conv, V, out);
}